// FSdrnnCorr_63668595196267
// MI455X (gfx1250) — compile-verified
//
#include <hip/hip_runtime.h>
#include <hip/hip_bf16.h>
#include <math.h>

typedef __attribute__((ext_vector_type(2))) float v2f;
typedef __attribute__((ext_vector_type(4))) float v4f;
typedef __attribute__((ext_vector_type(8))) float v8f;

#define B_TOT   32768
#define IN_DIM  2048
#define HID     128
#define NV      20
#define NP      6
#define NR      10

#define KC      32      // K chunk staged through LDS
#define PITCH   36      // 32 + 4 pad: conflict-free AND 16B-aligned rows

// ---------------------------------------------------------------------------
// Kernel A: z = relu(x @ We1 + be1) @ We2 + be2   (fused; hidden never stored)
// Block: 256 threads = 8 waves. Block tile: 128 rows x 128 cols. Wave: 16x128.
// Double-buffered LDS: global loads of chunk k+1 overlap the 64 WMMAs of k.
// ---------------------------------------------------------------------------
__global__ __launch_bounds__(256)
void encoder_kernel(const float* __restrict__ x,
                    const float* __restrict__ We1,
                    const float* __restrict__ be1,
                    const float* __restrict__ We2,
                    const float* __restrict__ be2,
                    float* __restrict__ zout)
{
    __shared__ __align__(16) float x_lds[2][128 * PITCH];   // [row][k]
    __shared__ __align__(16) float w_lds[2][HID * PITCH];   // [col][k] (transposed)

    const int tid  = threadIdx.x;
    const int wave = tid >> 5;
    const int lane = tid & 31;
    const int ln   = lane & 15;
    const int half = lane >> 4;
    const int koff = half << 1;            // K offset per half-wave (A/B frag layout)

    const int rowBase = blockIdx.x * 128;

    // ---- staging index precompute (constant strides -> immediate offsets) ----
    // x chunk: 128 rows x 32 k as 1024 float4; thread does 4, row stride 32.
    const int xr0 = tid >> 3;              // 0..31
    const int xc4 = (tid & 7) << 2;        // 0,4,...,28
    const float* xg0 = &x[(size_t)(rowBase + xr0) * IN_DIM + xc4];
    // We1 chunk transposed: col cc, 16 consecutive k per thread.
    const int wcc = tid & 127;
    const int wr0 = (tid >> 7) << 4;       // 0 or 16
    const float* wg0 = &We1[(size_t)wr0 * HID + wcc];

    v8f c[8];
    #pragma unroll
    for (int t = 0; t < 8; ++t)
        #pragma unroll
        for (int e = 0; e < 8; ++e) c[t][e] = 0.0f;

    v4f  xt[4];
    float wt[16];

    // ---- prologue: load + store chunk 0 ----
    #pragma unroll
    for (int i = 0; i < 4; ++i)
        xt[i] = __builtin_nontemporal_load(
            reinterpret_cast<const v4f*>(xg0 + (size_t)i * 32 * IN_DIM));
    #pragma unroll
    for (int i = 0; i < 16; ++i)
        wt[i] = wg0[(size_t)i * HID];
    #pragma unroll
    for (int i = 0; i < 4; ++i)
        *reinterpret_cast<v4f*>(&x_lds[0][(xr0 + i * 32) * PITCH + xc4]) = xt[i];
    #pragma unroll
    for (int j = 0; j < 4; ++j) {
        v4f wv = {wt[4 * j], wt[4 * j + 1], wt[4 * j + 2], wt[4 * j + 3]};
        *reinterpret_cast<v4f*>(&w_lds[0][wcc * PITCH + wr0 + 4 * j]) = wv;
    }
    __syncthreads();

    int cur = 0;
    for (int kk = 0; kk < IN_DIM; kk += KC) {
        const int  nxt      = cur ^ 1;
        const bool has_next = (kk + KC) < IN_DIM;

        // Issue next chunk's global loads NOW; waits land after the WMMA block.
        if (has_next) {
            const float* xg = xg0 + (kk + KC);
            const float* wg = wg0 + (size_t)(kk + KC) * HID;
            #pragma unroll
            for (int i = 0; i < 4; ++i)
                xt[i] = __builtin_nontemporal_load(
                    reinterpret_cast<const v4f*>(xg + (size_t)i * 32 * IN_DIM));
            #pragma unroll
            for (int i = 0; i < 16; ++i)
                wt[i] = wg[(size_t)i * HID];
        }

        // ---- compute on current buffer: 8 k-steps x 8 col-tiles ----
        const float* xb = &x_lds[cur][(wave * 16 + ln) * PITCH];
        const float* wb = &w_lds[cur][ln * PITCH];
        #pragma unroll
        for (int ks = 0; ks < 8; ++ks) {
            const int kl = (ks << 2) + koff;      // 4*ks + {0,2} per half-wave
            const v2f a = *reinterpret_cast<const v2f*>(xb + kl);
            v2f bf[8];
            #pragma unroll
            for (int t = 0; t < 8; ++t)
                bf[t] = *reinterpret_cast<const v2f*>(wb + t * 16 * PITCH + kl);
            #pragma unroll
            for (int t = 0; t < 8; ++t)
                c[t] = __builtin_amdgcn_wmma_f32_16x16x4_f32(
                    false, a, false, bf[t], (short)0, c[t], false, false);
        }

        // ---- drain loads into the other buffer ----
        if (has_next) {
            #pragma unroll
            for (int i = 0; i < 4; ++i)
                *reinterpret_cast<v4f*>(&x_lds[nxt][(xr0 + i * 32) * PITCH + xc4]) = xt[i];
            #pragma unroll
            for (int j = 0; j < 4; ++j) {
                v4f wv = {wt[4 * j], wt[4 * j + 1], wt[4 * j + 2], wt[4 * j + 3]};
                *reinterpret_cast<v4f*>(&w_lds[nxt][wcc * PITCH + wr0 + 4 * j]) = wv;
            }
        }
        __syncthreads();
        cur = nxt;
    }

    // ---- epilogue: y = relu(C + be1); z = y @ We2 + be2 (reduce over 16 lanes) ----
    float be1v[8], w20[8], w21[8];
    #pragma unroll
    for (int t = 0; t < 8; ++t) {
        int col = t * 16 + ln;
        be1v[t] = be1[col];
        float2 w = *reinterpret_cast<const float2*>(&We2[col * 2]);
        w20[t] = w.x; w21[t] = w.y;
    }
    const float bz0 = be2[0], bz1 = be2[1];

    #pragma unroll
    for (int vi = 0; vi < 8; ++vi) {
        float a0 = 0.0f, a1 = 0.0f;
        #pragma unroll
        for (int t = 0; t < 8; ++t) {
            float y = fmaxf(c[t][vi] + be1v[t], 0.0f);
            a0 = fmaf(y, w20[t], a0);
            a1 = fmaf(y, w21[t], a1);
        }
        #pragma unroll
        for (int m = 1; m < 16; m <<= 1) {     // reduce N dim (stays in half-wave)
            a0 += __shfl_xor(a0, m, 32);
            a1 += __shfl_xor(a1, m, 32);
        }
        if (ln == 0) {
            int row = rowBase + wave * 16 + half * 8 + vi;  // C layout: M = 8*half + v
            float2 zv; zv.x = a0 + bz0; zv.y = a1 + bz1;
            *reinterpret_cast<float2*>(&zout[row * 2]) = zv;
        }
    }
}

// ---------------------------------------------------------------------------
// Kernel B: per-head MLP + LoRA + double tanh. Grid (B/256, V). Weights in LDS.
// ---------------------------------------------------------------------------
__global__ __launch_bounds__(256)
void heads_kernel(const float* __restrict__ z,
                  const float* __restrict__ W1,
                  const float* __restrict__ b1,
                  const float* __restrict__ W2,
                  const float* __restrict__ b2,
                  const float* __restrict__ lora_A,
                  const float* __restrict__ lora_B,
                  float* __restrict__ out)
{
    const int v   = blockIdx.y;
    const int tid = threadIdx.x;

    __shared__ float w1s[2 * HID];      // [d][c]
    __shared__ float b1s[HID];
    __shared__ float w2s[HID * NP];     // [c][p]
    __shared__ float b2s[NP];
    __shared__ float ms[2 * NP];        // fused lora_A @ lora_B cols for head v

    w1s[tid] = W1[(size_t)v * 2 * HID + tid];
    if (tid < HID) b1s[tid] = b1[v * HID + tid];
    #pragma unroll
    for (int i = 0; i < 3; ++i) {
        int e = tid + 256 * i;
        w2s[e] = W2[(size_t)v * HID * NP + e];
    }
    if (tid < NP) b2s[tid] = b2[v * NP + tid];
    if (tid < 2 * NP) {
        int d = tid / NP, p = tid % NP;
        float s = 0.0f;
        #pragma unroll
        for (int r = 0; r < NR; ++r)
            s = fmaf(lora_A[d * NR + r], lora_B[r * (NV * NP) + v * NP + p], s);
        ms[tid] = s;
    }
    __syncthreads();

    const int b = blockIdx.x * 256 + tid;
    const float2 zz = *reinterpret_cast<const float2*>(&z[(size_t)b * 2]);

    float acc[NP];
    #pragma unroll
    for (int p = 0; p < NP; ++p) acc[p] = b2s[p];

    for (int cidx = 0; cidx < HID; ++cidx) {               // broadcast LDS reads
        float hc = fmaf(zz.x, w1s[cidx], fmaf(zz.y, w1s[HID + cidx], b1s[cidx]));
        hc = fmaxf(hc, 0.0f);
        #pragma unroll
        for (int p = 0; p < NP; ++p)
            acc[p] = fmaf(hc, w2s[cidx * NP + p], acc[p]);
    }

    float* o = &out[(size_t)b * (NV * NP) + v * NP];       // 6 contiguous floats
    #pragma unroll
    for (int p = 0; p < NP; ++p) {
        float lora = fmaf(zz.x, ms[p], zz.y * ms[NP + p]);
        __builtin_nontemporal_store(tanhf(tanhf(acc[p]) + 0.15f * lora), &o[p]);
    }
}

// ---------------------------------------------------------------------------
extern "C" void kernel_launch(void* const* d_in, const int* in_sizes, int n_in,
                              void* d_out, int out_size, void* d_ws, size_t ws_size,
                              hipStream_t stream) {
    const float* x    = (const float*)d_in[0];
    const float* We1  = (const float*)d_in[1];
    const float* be1  = (const float*)d_in[2];
    const float* We2  = (const float*)d_in[3];
    const float* be2  = (const float*)d_in[4];
    const float* W1   = (const float*)d_in[5];
    const float* b1   = (const float*)d_in[6];
    const float* W2   = (const float*)d_in[7];
    const float* b2   = (const float*)d_in[8];
    const float* lA   = (const float*)d_in[9];
    const float* lB   = (const float*)d_in[10];

    float* out  = (float*)d_out;
    float* zout = out + (size_t)B_TOT * NV * NP;   // z section of tuple output

    encoder_kernel<<<dim3(B_TOT / 128), 256, 0, stream>>>(x, We1, be1, We2, be2, zout);
    heads_kernel<<<dim3(B_TOT / 256, NV), 256, 0, stream>>>(zout, W1, b1, W2, b2, lA, lB, out);
}